// GraphAttention_56762287784161
// MI455X (gfx1250) — compile-verified
//
#include <hip/hip_runtime.h>
#include <hip/hip_bf16.h>

// ---------------------------------------------------------------------------
// GAT layer, fused for MI455X (gfx1250, wave32, WMMA + TDM).
// Shapes: B=4, N=2048, F_IN=256, H=8, D=32.
//
// Kernel 1: Wh = x @ W (per head) via V_WMMA_F32_16X16X4_F32, plus
//           s1 = Wh@a1, s2 = Wh@a2 via wave32 xor-shuffle row reductions.
// Kernel 2: flash-attention style masked softmax + (P @ Wh) via WMMA.
//           adj tiles are DMA'd into double-buffered LDS by the Tensor Data
//           Mover (tensor_load_to_lds + s_wait_tensorcnt), shared by 8 heads.
// ---------------------------------------------------------------------------

#define B_    4
#define N_    2048
#define FIN_  256
#define H_    8
#define D_    32
#define ALPHA_    0.2f
#define MASK_VAL_ (-9000000000000000.0f)

typedef __attribute__((ext_vector_type(2))) float        v2f;
typedef __attribute__((ext_vector_type(8))) float        v8f;
typedef __attribute__((ext_vector_type(4))) unsigned int u32x4;
typedef __attribute__((ext_vector_type(8))) int          i32x8;
typedef __attribute__((ext_vector_type(4))) int          i32x4;

#if __has_builtin(__builtin_amdgcn_tensor_load_to_lds) && \
    __has_builtin(__builtin_amdgcn_s_wait_tensorcnt)
#define GAT_HAVE_TDM 1
#else
#define GAT_HAVE_TDM 0
#endif

// ---------------------------------------------------------------------------
// Kernel 1: projection + attention-source scores.
// grid = B*H*(N/128) blocks of 256 threads (8 waves; each wave: 16 rows).
// ---------------------------------------------------------------------------
__global__ __launch_bounds__(256) void gat_proj_kernel(
    const float* __restrict__ x,   // [B][N][FIN]
    const float* __restrict__ W,   // [H][FIN][D]
    const float* __restrict__ a,   // [H][2D]
    float* __restrict__ Wh,        // [B*H][N][D]   (workspace)
    float* __restrict__ s1,        // [B*H][N]      (workspace)
    float* __restrict__ s2)        // [B*H][N]      (workspace)
{
    __shared__ __align__(16) float Wlds[FIN_ * D_];  // 32 KB: W for this head

    const int bh     = blockIdx.x >> 4;        // N_/128 == 16 row-slabs
    const int rowBlk = blockIdx.x & 15;
    const int b      = bh / H_;
    const int h      = bh % H_;

    // Cooperative load of W_h into LDS (2048 float4 by 256 threads).
    {
        const float4* src = (const float4*)(W + (size_t)h * FIN_ * D_);
        float4*       dst = (float4*)Wlds;
#pragma unroll
        for (int i = 0; i < 8; ++i)
            dst[threadIdx.x + i * 256] = src[threadIdx.x + i * 256];
    }
    __syncthreads();

    const int wave = threadIdx.x >> 5;
    const int lane = threadIdx.x & 31;
    const int hl   = lane >> 4;       // half of wave (0/1)
    const int lm   = lane & 15;

    const int i0   = rowBlk * 128 + wave * 16;
    const int row  = i0 + lm;                       // A-matrix row (M = lane%16)
    const float* xrow = x + ((size_t)b * N_ + row) * FIN_;

    v8f c0 = {};  // cols  0..15
    v8f c1 = {};  // cols 16..31

    // K loop: 256 in chunks of 4 (one WMMA_F32_16X16X4_F32 per chunk per tile).
#pragma unroll 4
    for (int kc = 0; kc < FIN_ / 4; ++kc) {
        const int k = kc * 4 + 2 * hl;             // A/B VGPR v holds K = v + 2*half
        v2f A = *(const v2f*)(xrow + k);           // A[M=lm][k], A[M=lm][k+1]
        v2f Bv0, Bv1;
        Bv0.x = Wlds[k * D_ + lm];
        Bv0.y = Wlds[(k + 1) * D_ + lm];
        Bv1.x = Wlds[k * D_ + 16 + lm];
        Bv1.y = Wlds[(k + 1) * D_ + 16 + lm];
        c0 = __builtin_amdgcn_wmma_f32_16x16x4_f32(false, A, false, Bv0,
                                                   (short)0, c0, false, false);
        c1 = __builtin_amdgcn_wmma_f32_16x16x4_f32(false, A, false, Bv1,
                                                   (short)0, c1, false, false);
    }

    // a1/a2 slices for this lane's two column positions.
    const float* ah = a + h * 2 * D_;
    const float a1a = ah[lm],        a1b = ah[16 + lm];
    const float a2a = ah[D_ + lm],   a2b = ah[D_ + 16 + lm];

    float* WhRow = Wh + ((size_t)bh * N_ + i0) * D_;

#pragma unroll
    for (int v = 0; v < 8; ++v) {
        const int r = v + 8 * hl;                  // C-layout: row = v + 8*half
        // Store Wh tile.
        WhRow[(size_t)r * D_ + lm]      = c0[v];
        WhRow[(size_t)r * D_ + 16 + lm] = c1[v];
        // Row-dot with a1/a2: xor-shuffle reduce across the 16 lanes of a half.
        float p1 = c0[v] * a1a + c1[v] * a1b;
        float p2 = c0[v] * a2a + c1[v] * a2b;
#pragma unroll
        for (int off = 1; off < 16; off <<= 1) {
            p1 += __shfl_xor(p1, off, 32);
            p2 += __shfl_xor(p2, off, 32);
        }
        if (lm == 0) {
            s1[(size_t)bh * N_ + i0 + r] = p1;
            s2[(size_t)bh * N_ + i0 + r] = p2;
        }
    }
}

#if GAT_HAVE_TDM
// Issue one TDM DMA: 16x16 i32 tile of adj (tile start = gptr) into LDS at byte offset ldsOff.
// D#: 2D tensor, data_size=4B, tensor_dim0/1 = N_, row stride = N_ elements.
__device__ __forceinline__ void tdm_load_adj_tile(const int* gptr,
                                                  unsigned   ldsOff) {
    const unsigned long long ga = (unsigned long long)(uintptr_t)gptr;
    u32x4 g0;
    g0.x = 1u;                                            // count=1 (user D#)
    g0.y = ldsOff;                                        // lds_addr (bytes)
    g0.z = (unsigned)ga;                                  // global_addr[31:0]
    g0.w = (unsigned)((ga >> 32) & 0x01FFFFFFu)           // global_addr[56:32]
           | (2u << 30);                                  // type = 2 ("image")
    i32x8 g1;
    g1[0] = (2 << 16);                                    // data_size = 4B
    g1[1] = (int)((unsigned)N_ << 16);                    // tensor_dim0[15:0]
    g1[2] = (int)((unsigned)N_ << 16);                    // dim0 hi | tensor_dim1 lo
    g1[3] = (16 << 16);                                   // dim1 hi | tile_dim0=16
    g1[4] = 16;                                           // tile_dim1=16, tile_dim2=0
    g1[5] = N_;                                           // tensor_dim0_stride lo
    g1[6] = 0;                                            // stride hi | dim1_stride lo
    g1[7] = 0;
    i32x4 gz4 = {0, 0, 0, 0};                             // 2D: groups 2/3 unused
#if defined(__clang_major__) && __clang_major__ >= 23
    i32x8 gz8 = {0, 0, 0, 0, 0, 0, 0, 0};
    __builtin_amdgcn_tensor_load_to_lds(g0, g1, gz4, gz4, gz8, 0);
#else
    __builtin_amdgcn_tensor_load_to_lds(g0, g1, gz4, gz4, 0);
#endif
}
#endif  // GAT_HAVE_TDM

// ---------------------------------------------------------------------------
// Kernel 2: fused masked-softmax attention + AV product (online softmax).
// grid = B*(N/16) blocks of 256 threads; wave w handles head w.
// adj 16x16 tiles are double-buffered in LDS via the Tensor Data Mover and
// shared by all 8 heads (8x reduction of adj traffic + DMA/compute overlap).
// ---------------------------------------------------------------------------
__global__ __launch_bounds__(256) void gat_attn_kernel(
    const int*   __restrict__ adj,  // [B][N][N]
    const float* __restrict__ Wh,   // [B*H][N][D]
    const float* __restrict__ s1,   // [B*H][N]
    const float* __restrict__ s2,   // [B*H][N]
    float* __restrict__ out)        // [B][N][H*D]
{
    __shared__ int adjbuf[2][16 * 16];                // double-buffered adj tile
    __shared__ __align__(16) float Pbuf[H_][16 * 16]; // per-wave P transpose

    const int b  = blockIdx.x / (N_ / 16);
    const int iT = blockIdx.x % (N_ / 16);
    const int i0 = iT * 16;

    const int w    = threadIdx.x >> 5;   // wave id == head id
    const int lane = threadIdx.x & 31;
    const int hl   = lane >> 4;
    const int lm   = lane & 15;
    const int bh   = b * H_ + w;

    const float* s1p = s1 + (size_t)bh * N_;
    const float* s2p = s2 + (size_t)bh * N_;
    const float* V   = Wh + (size_t)bh * N_ * D_;

    // Per-lane query-row scores (C-layout rows: v + 8*half).
    float s1v[8];
#pragma unroll
    for (int v = 0; v < 8; ++v) s1v[v] = s1p[i0 + v + 8 * hl];

    float m[8], ell[8];
#pragma unroll
    for (int v = 0; v < 8; ++v) { m[v] = -INFINITY; ell[v] = 0.0f; }
    v8f acc0 = {};
    v8f acc1 = {};

    const int*   adjTileBase = adj + ((size_t)b * N_ + i0) * N_;
    const int    NJT         = N_ / 16;

#if GAT_HAVE_TDM
    // Prologue: wave 0 kicks off the DMA for tile 0 (TDM ignores EXEC; one
    // wave-uniform issue per workgroup).
    if (w == 0) {
        tdm_load_adj_tile(adjTileBase, (unsigned)(uintptr_t)&adjbuf[0][0]);
    }
#endif

    for (int jt = 0; jt < NJT; ++jt) {
        const int j0  = jt * 16;
        const int cur = jt & 1;

#if GAT_HAVE_TDM
        __syncthreads();  // buf[1-cur] reads from iteration jt-1 complete
        if (w == 0) {
            if (jt + 1 < NJT) {
                // Overlap: issue tile jt+1 into the freed buffer, then wait
                // for the *oldest* outstanding DMA (tile jt) only.
                tdm_load_adj_tile(adjTileBase + (size_t)(j0 + 16),
                                  (unsigned)(uintptr_t)&adjbuf[1 - cur][0]);
                __builtin_amdgcn_s_wait_tensorcnt((short)1);
            } else {
                __builtin_amdgcn_s_wait_tensorcnt((short)0);
            }
        }
        __syncthreads();  // buf[cur] now visible to all 8 waves
        const int* adjT = adjbuf[cur];
#else
        __syncthreads();
        adjbuf[0][threadIdx.x] =
            adjTileBase[(size_t)(threadIdx.x >> 4) * N_ + j0 + (threadIdx.x & 15)];
        if ((jt + 1) < NJT && threadIdx.x < 16)
            __builtin_prefetch(&adjTileBase[(size_t)threadIdx.x * N_ + j0 + 16], 0, 0);
        __syncthreads();
        const int* adjT = adjbuf[0];
#endif

        const float s2l = s2p[j0 + lm];

        float pv[8];
#pragma unroll
        for (int v = 0; v < 8; ++v) {
            const int r = v + 8 * hl;
            float e = s1v[v] + s2l;
            e = (e > 0.0f) ? e : ALPHA_ * e;                // leaky relu
            e = adjT[r * 16 + lm] ? e : MASK_VAL_;          // mask
            // row max (16-lane xor reduce within each half)
            float rm = e;
#pragma unroll
            for (int off = 1; off < 16; off <<= 1)
                rm = fmaxf(rm, __shfl_xor(rm, off, 32));
            const float mn = fmaxf(m[v], rm);
            const float sc = __expf(m[v] - mn);
            m[v] = mn;
            const float p = __expf(e - mn);
            float rs = p;
#pragma unroll
            for (int off = 1; off < 16; off <<= 1)
                rs += __shfl_xor(rs, off, 32);
            ell[v]  = ell[v] * sc + rs;
            acc0[v] *= sc;
            acc1[v] *= sc;
            pv[v] = p;
        }

        // Transpose P: C-layout -> A-layout via this wave's LDS slab.
#pragma unroll
        for (int v = 0; v < 8; ++v)
            Pbuf[w][(v + 8 * hl) * 16 + lm] = pv[v];
        asm volatile("s_wait_dscnt 0" ::: "memory");  // intra-wave LDS RAW

        // P(16x16) @ V(16x32): 4 K-chunks x 2 N-tiles of WMMA_F32_16X16X4.
#pragma unroll
        for (int kc = 0; kc < 4; ++kc) {
            const int kk = kc * 4 + 2 * hl;           // A/B VGPR v: K = v + 2*half
            v2f Af = *(const v2f*)&Pbuf[w][lm * 16 + kk];
            const float* vr = V + (size_t)(j0 + kk) * D_;
            v2f Bv0, Bv1;
            Bv0.x = vr[lm];          Bv0.y = vr[D_ + lm];
            Bv1.x = vr[16 + lm];     Bv1.y = vr[D_ + 16 + lm];
            acc0 = __builtin_amdgcn_wmma_f32_16x16x4_f32(false, Af, false, Bv0,
                                                         (short)0, acc0, false, false);
            acc1 = __builtin_amdgcn_wmma_f32_16x16x4_f32(false, Af, false, Bv1,
                                                         (short)0, acc1, false, false);
        }
    }

    // Epilogue: normalize, ELU, scatter to [B][N][H*D].
#pragma unroll
    for (int v = 0; v < 8; ++v) {
        const float inv = 1.0f / ell[v];
        const int   r   = i0 + v + 8 * hl;
        float o0 = acc0[v] * inv;
        float o1 = acc1[v] * inv;
        o0 = (o0 > 0.0f) ? o0 : (__expf(o0) - 1.0f);
        o1 = (o1 > 0.0f) ? o1 : (__expf(o1) - 1.0f);
        float* orow = out + ((size_t)b * N_ + r) * (H_ * D_) + w * D_;
        orow[lm]      = o0;
        orow[16 + lm] = o1;
    }
}

// ---------------------------------------------------------------------------
extern "C" void kernel_launch(void* const* d_in, const int* in_sizes, int n_in,
                              void* d_out, int out_size, void* d_ws, size_t ws_size,
                              hipStream_t stream) {
    const float* x   = (const float*)d_in[0];  // [4][2048][256] f32
    const int*   adj = (const int*)  d_in[1];  // [4][2048][2048] i32
    const float* W   = (const float*)d_in[2];  // [8][256][32] f32
    const float* a   = (const float*)d_in[3];  // [8][64][1] f32
    float*       out = (float*)d_out;          // [4][2048][256] f32

    // Workspace layout: Wh (8 MB) | s1 (256 KB) | s2 (256 KB)
    float* Wh = (float*)d_ws;
    float* s1 = Wh + (size_t)B_ * H_ * N_ * D_;
    float* s2 = s1 + (size_t)B_ * H_ * N_;

    gat_proj_kernel<<<B_ * H_ * (N_ / 128), 256, 0, stream>>>(x, W, a, Wh, s1, s2);
    gat_attn_kernel<<<B_ * (N_ / 16),       256, 0, stream>>>(adj, Wh, s1, s2, out);
}